// AdditiveAttention_36721970381298
// MI455X (gfx1250) — compile-verified
//
#include <hip/hip_runtime.h>

#define B_ 8
#define Q_ 128
#define K_ 1024
#define D_ 256
#define H_ 256
#define KCH 32
#define QB 4

typedef __attribute__((ext_vector_type(2))) float v2f;
typedef __attribute__((ext_vector_type(8))) float v8f;

// ---------------------------------------------------------------------------
// Kernel 1: projection GEMM  P[b,m,h] = sum_d X[b,m,d] * W[h,d]
// One wave per 16x16 (m,h) output tile, V_WMMA_F32_16X16X4_F32 over D (64 steps).
// grid.x = B*(M/16), grid.y = H/64, block = 128 (4 waves)
// ---------------------------------------------------------------------------
__global__ void proj_gemm_kernel(const float* __restrict__ X,
                                 const float* __restrict__ W,
                                 float* __restrict__ P, int M) {
  const int lane = threadIdx.x & 31;
  const int wave = threadIdx.x >> 5;
  const int mtiles = M >> 4;
  const int b  = blockIdx.x / mtiles;
  const int m0 = (blockIdx.x % mtiles) << 4;
  const int h0 = (blockIdx.y * 4 + wave) << 4;
  const int rc   = lane & 15;          // A row / B col / C col
  const int koff = (lane >> 4) << 1;   // K sub-offset: 0 or 2

  const float* xrow = X + (size_t)(b * M + m0 + rc) * D_;
  const float* wrow = W + (size_t)(h0 + rc) * D_;

  v8f acc = {};
  for (int d0 = 0; d0 < D_; d0 += 4) {
    v2f a  = *reinterpret_cast<const v2f*>(xrow + d0 + koff);  // 8B aligned
    v2f bb = *reinterpret_cast<const v2f*>(wrow + d0 + koff);
    acc = __builtin_amdgcn_wmma_f32_16x16x4_f32(false, a, false, bb,
                                                (short)0, acc, false, false);
  }
  const int mbase = m0 + ((lane >> 4) << 3);
  float* pcol = P + (size_t)(b * M) * H_ + h0 + rc;
#pragma unroll
  for (int i = 0; i < 8; ++i)
    pcol[(size_t)(mbase + i) * H_] = acc[i];
}

// ---------------------------------------------------------------------------
// Kernel 2: scores[b,q,k] = sum_h w_v[h] * tanh(qproj[b,q,h] + kproj[b,k,h]),
// masked to -1e9 for k >= valid_len[b].  Bottleneck stage: 268M v_tanh_f32.
//  - qproj rows (QB=4 per block) and w_v held in REGISTERS (zero LDS reads)
//  - kproj tile (32 x 256 f32 = 32KB) staged in LDS via float4
//  - per k: 8 LDS reads feed 32 tanh (4 q's reuse the same krow registers)
// grid.x = B*Q/QB, grid.y = K/KCH, block = 256 (8 waves; each wave -> 4 k's)
// ---------------------------------------------------------------------------
__global__ void scores_kernel(const float* __restrict__ qproj,
                              const float* __restrict__ kproj,
                              const float* __restrict__ wv,
                              const int* __restrict__ valid_lens,
                              float* __restrict__ scores) {
  __shared__ float4 sk4[(KCH * H_) / 4];          // 32 KB
  float* sk = (float*)sk4;

  const int qtiles = Q_ / QB;                     // 32
  const int b   = blockIdx.x / qtiles;
  const int q0  = (blockIdx.x % qtiles) * QB;
  const int k0  = blockIdx.y * KCH;
  const int tid  = threadIdx.x;
  const int wave = tid >> 5;
  const int lane = tid & 31;

  // registers: 4 qproj rows (8 h-values per lane) + w_v
  float qv[QB][8];
  float wvv[8];
#pragma unroll
  for (int s = 0; s < 8; ++s) {
    wvv[s] = wv[lane + (s << 5)];
#pragma unroll
    for (int qq = 0; qq < QB; ++qq)
      qv[qq][s] = qproj[(size_t)(b * Q_ + q0 + qq) * H_ + lane + (s << 5)];
  }

  // cooperative 32KB LDS fill, b128 granularity (8 iters per thread)
  const float4* kp4 =
      (const float4*)(kproj + ((size_t)b * K_ + k0) * H_);
  for (int i = tid; i < (KCH * H_) / 4; i += 256) sk4[i] = kp4[i];

  const int vlen = valid_lens[b];
  __syncthreads();

#pragma unroll
  for (int j = 0; j < 4; ++j) {
    const int kk = (wave << 2) + j;
    const float* krow = sk + kk * H_;
    float kr[8];
#pragma unroll
    for (int s = 0; s < 8; ++s) kr[s] = krow[lane + (s << 5)];

    const int kg = k0 + kk;
#pragma unroll
    for (int qq = 0; qq < QB; ++qq) {
      float acc = 0.f;
#pragma unroll
      for (int s = 0; s < 8; s += 4) {
        float x0 = qv[qq][s + 0] + kr[s + 0];
        float x1 = qv[qq][s + 1] + kr[s + 1];
        float x2 = qv[qq][s + 2] + kr[s + 2];
        float x3 = qv[qq][s + 3] + kr[s + 3];
        float t0, t1, t2, t3;
        // 4 independent trans ops back-to-back: each covers the previous
        // one's 1-op output hazard; trailing v_nop covers the last.
        asm volatile("v_tanh_f32 %0, %4\n\t"
                     "v_tanh_f32 %1, %5\n\t"
                     "v_tanh_f32 %2, %6\n\t"
                     "v_tanh_f32 %3, %7\n\t"
                     "v_nop"
                     : "=&v"(t0), "=&v"(t1), "=&v"(t2), "=&v"(t3)
                     : "v"(x0), "v"(x1), "v"(x2), "v"(x3));
        acc = fmaf(wvv[s + 0], t0, acc);
        acc = fmaf(wvv[s + 1], t1, acc);
        acc = fmaf(wvv[s + 2], t2, acc);
        acc = fmaf(wvv[s + 3], t3, acc);
      }
#pragma unroll
      for (int off = 16; off > 0; off >>= 1)
        acc += __shfl_xor(acc, off, 32);
      if (lane == 0) {
        scores[(size_t)(b * Q_ + q0 + qq) * K_ + kg] =
            (kg < vlen) ? acc : -1e9f;
      }
    }
  }
}

// ---------------------------------------------------------------------------
// Kernel 3: row softmax over K=1024 (mask already applied as -1e9).
// grid = B*Q, block = 256; each thread owns 4 strided elements.
// ---------------------------------------------------------------------------
__global__ void softmax_kernel(float* __restrict__ scores) {
  const int bq = blockIdx.x;
  float* row = scores + (size_t)bq * K_;
  const int tid  = threadIdx.x;
  const int wave = tid >> 5, lane = tid & 31;
  __shared__ float smax[8];
  __shared__ float ssum[8];

  float x[4];
  float m = -3.402823466e38f;
#pragma unroll
  for (int i = 0; i < 4; ++i) { x[i] = row[tid + (i << 8)]; m = fmaxf(m, x[i]); }
#pragma unroll
  for (int off = 16; off > 0; off >>= 1) m = fmaxf(m, __shfl_xor(m, off, 32));
  if (lane == 0) smax[wave] = m;
  __syncthreads();
  if (tid < 8) {
    float t = smax[tid];
#pragma unroll
    for (int off = 4; off > 0; off >>= 1) t = fmaxf(t, __shfl_xor(t, off, 32));
    if (tid == 0) smax[0] = t;
  }
  __syncthreads();
  m = smax[0];

  float s = 0.f;
#pragma unroll
  for (int i = 0; i < 4; ++i) { x[i] = __expf(x[i] - m); s += x[i]; }
#pragma unroll
  for (int off = 16; off > 0; off >>= 1) s += __shfl_xor(s, off, 32);
  if (lane == 0) ssum[wave] = s;
  __syncthreads();
  if (tid < 8) {
    float t = ssum[tid];
#pragma unroll
    for (int off = 4; off > 0; off >>= 1) t += __shfl_xor(t, off, 32);
    if (tid == 0) ssum[0] = t;
  }
  __syncthreads();
  const float inv = 1.f / ssum[0];
#pragma unroll
  for (int i = 0; i < 4; ++i) row[tid + (i << 8)] = x[i] * inv;
}

// ---------------------------------------------------------------------------
// Kernel 4: out[b,q,v] = sum_k attn[b,q,k] * values[b,k,v]   (fp32 WMMA)
// grid.x = B*(Q/16), grid.y = D/64, block = 128 (4 waves), 256 WMMAs per tile
// ---------------------------------------------------------------------------
__global__ void av_gemm_kernel(const float* __restrict__ attn,
                               const float* __restrict__ V,
                               float* __restrict__ out) {
  const int lane = threadIdx.x & 31;
  const int wave = threadIdx.x >> 5;
  const int b  = blockIdx.x >> 3;            // Q/16 = 8 tiles
  const int q0 = (blockIdx.x & 7) << 4;
  const int v0 = (blockIdx.y * 4 + wave) << 4;
  const int rc   = lane & 15;
  const int koff = (lane >> 4) << 1;

  const float* arow  = attn + (size_t)(b * Q_ + q0 + rc) * K_;
  const float* vbase = V + (size_t)b * K_ * D_ + v0 + rc;

  v8f acc = {};
  for (int k0 = 0; k0 < K_; k0 += 4) {
    v2f a = *reinterpret_cast<const v2f*>(arow + k0 + koff);
    v2f bb;
    bb.x = vbase[(size_t)(k0 + koff + 0) * D_];   // lanes 0-15: coalesced in v
    bb.y = vbase[(size_t)(k0 + koff + 1) * D_];
    acc = __builtin_amdgcn_wmma_f32_16x16x4_f32(false, a, false, bb,
                                                (short)0, acc, false, false);
  }
  const int qbase = q0 + ((lane >> 4) << 3);
  float* ocol = out + (size_t)b * Q_ * D_ + v0 + rc;
#pragma unroll
  for (int i = 0; i < 8; ++i)
    ocol[(size_t)(qbase + i) * D_] = acc[i];
}

// ---------------------------------------------------------------------------
extern "C" void kernel_launch(void* const* d_in, const int* in_sizes, int n_in,
                              void* d_out, int out_size, void* d_ws, size_t ws_size,
                              hipStream_t stream) {
  const float* queries = (const float*)d_in[0];   // [B,Q,D]
  const float* keys    = (const float*)d_in[1];   // [B,K,D]
  const float* values  = (const float*)d_in[2];   // [B,K,D]
  const float* W_q     = (const float*)d_in[3];   // [H,D]
  const float* W_k     = (const float*)d_in[4];   // [H,D]
  const float* w_v     = (const float*)d_in[5];   // [H]
  const int*   vlens   = (const int*)d_in[6];     // [B]
  float* out = (float*)d_out;                     // [B,Q,D]

  float* ws     = (float*)d_ws;
  float* qproj  = ws;                                   // B*Q*H  = 256K f32
  float* kproj  = qproj + (size_t)B_ * Q_ * H_;         // B*K*H  = 2M   f32
  float* scores = kproj + (size_t)B_ * K_ * H_;         // B*Q*K  = 1M   f32

  proj_gemm_kernel<<<dim3(B_ * (Q_ / 16), H_ / 64), 128, 0, stream>>>(
      queries, W_q, qproj, Q_);
  proj_gemm_kernel<<<dim3(B_ * (K_ / 16), H_ / 64), 128, 0, stream>>>(
      keys, W_k, kproj, K_);
  scores_kernel<<<dim3(B_ * (Q_ / QB), K_ / KCH), 256, 0, stream>>>(
      qproj, kproj, w_v, vlens, scores);
  softmax_kernel<<<B_ * Q_, 256, 0, stream>>>(scores);
  av_gemm_kernel<<<dim3(B_ * (Q_ / 16), D_ / 64), 128, 0, stream>>>(
      scores, values, out);
}